// warpLayer_25950192403264
// MI455X (gfx1250) — compile-verified
//
#include <hip/hip_runtime.h>
#include <cmath>

// ---------------------------------------------------------------------------
// warpLayer for MI455X (gfx1250).
//
// Roofline: 4.19M pixels, 16B in + 48B out per pixel over HBM (~267MB total
// -> ~11.5us at 23.3 TB/s). The 2MB weight LUT is L2-resident (192MB L2), so
// the 128B/pixel of bilinear gathers never hit HBM. Memory-bound: the kernel
// is a straight stream with a small amount of VALU (2x atan2f + bilinear FMA)
// that overlaps. No matmul exists in this op, so WMMA is inapplicable; the
// CDNA5-specific path used instead is the async global->LDS data mover
// (GLOBAL_LOAD_ASYNC_TO_LDS_B128, ASYNCcnt) with double buffering so the next
// tile's image fetch overlaps the current tile's compute, plus non-temporal
// output stores so the 201MB write stream does not evict the hot LUT from L2.
// ---------------------------------------------------------------------------

typedef float v4f __attribute__((ext_vector_type(4)));
typedef int   v4i_ __attribute__((vector_size(16)));
typedef __attribute__((address_space(1))) v4i_* gv4_t;   // global int4*
typedef __attribute__((address_space(3))) v4i_* sv4_t;   // LDS int4*

#define PI_F 3.14159265358979323846f
// 255 / (2*pi)
#define ANG_SCALE_F 40.58451048843331f

constexpr int BLOCK = 256;  // 8 waves (wave32)
constexpr int TPB   = 8;    // tiles of 256 pixels per block

#if defined(__gfx1250__) && __has_builtin(__builtin_amdgcn_global_load_async_to_lds_b128)
#define HAS_ASYNC_LDS 1
#else
#define HAS_ASYNC_LDS 0
#endif

__device__ __forceinline__ void process_pixel(v4f v,
                                              const float* __restrict__ wgt,
                                              float* __restrict__ out,
                                              int pix) {
  // C = (pi - atan2(imgOdd, -imgEven)) * 255 / (2*pi), clamped to [0, 254]
  float c0 = (PI_F - atan2f(v.y, -v.x)) * ANG_SCALE_F;
  float c1 = (PI_F - atan2f(v.w, -v.z)) * ANG_SCALE_F;
  c0 = fminf(fmaxf(c0, 0.0f), 254.0f);
  c1 = fminf(fmaxf(c1, 0.0f), 254.0f);
  int i0 = (int)c0;            // trunc == floor since c >= 0
  int j0 = (int)c1;
  float w0 = c0 - (float)i0;
  float w1 = c1 - (float)j0;
  // fold the 5e-4 output scale into the bilinear coefficients
  float a00 = (1.0f - w0) * (1.0f - w1) * 5.0e-4f;
  float a10 = w0 * (1.0f - w1) * 5.0e-4f;
  float a01 = (1.0f - w0) * w1 * 5.0e-4f;
  float a11 = w0 * w1 * 5.0e-4f;

  // weight is [256][256][8] f32; row stride = 2048 floats = 512 v4f.
  // corner(0,0)=b[0..1], corner(0,1)=b[2..3] (contiguous 64B),
  // corner(1,0)=b[512..513], corner(1,1)=b[514..515]. All L2-resident.
  const v4f* __restrict__ b = (const v4f*)(wgt + ((i0 << 8) + j0) * 8);
  v4f c00lo = b[0],   c00hi = b[1];
  v4f c01lo = b[2],   c01hi = b[3];
  v4f c10lo = b[512], c10hi = b[513];
  v4f c11lo = b[514], c11hi = b[515];

  v4f r0 = c00lo * a00 + c01lo * a01 + c10lo * a10 + c11lo * a11;
  v4f r1 = c00hi * a00 + c01hi * a01 + c10hi * a10 + c11hi * a11;

  // 12 floats per pixel, 48B stride -> each v4f store is 16B aligned.
  v4f* o = (v4f*)(out + (size_t)pix * 12);
  __builtin_nontemporal_store(r0, o + 0);
  __builtin_nontemporal_store(r1, o + 1);
  __builtin_nontemporal_store(v,  o + 2);
}

__global__ __launch_bounds__(BLOCK) void warp_kernel(
    const float* __restrict__ img,   // [npix][4]
    const float* __restrict__ wgt,   // [256][256][8]
    float* __restrict__ out,         // [npix][12]
    int npix) {
  const int tid = threadIdx.x;
  const int num_tiles = (npix + BLOCK - 1) / BLOCK;
  const int t0 = blockIdx.x * TPB;
  const int tend = min(t0 + TPB, num_tiles);
  if (t0 >= num_tiles) return;

#if HAS_ASYNC_LDS
  // Double-buffered async image staging: each lane async-copies its own 16B
  // pixel into LDS (ASYNCcnt), overlapping the next tile's fetch with this
  // tile's compute. Loads complete in order, so asynccnt<=1 releases the
  // older of two outstanding tile copies.
  __shared__ v4f lbuf[2][BLOCK];
#define ISSUE_TILE(bufi, tile)                                                \
  __builtin_amdgcn_global_load_async_to_lds_b128(                             \
      (gv4_t)(void*)(((const v4f*)img) + (size_t)(tile) * BLOCK + tid),       \
      (sv4_t)(void*)&lbuf[(bufi)][tid], 0, 0)

  ISSUE_TILE(0, t0);
  for (int k = 0; k < TPB; ++k) {
    const int tile = t0 + k;
    if (tile >= tend) break;
    if (tile + 1 < tend) {
      ISSUE_TILE((k + 1) & 1, tile + 1);
      asm volatile("s_wait_asynccnt 0x1" ::: "memory");
    } else {
      asm volatile("s_wait_asynccnt 0x0" ::: "memory");
    }
    v4f v = lbuf[k & 1][tid];
    const int pix = tile * BLOCK + tid;
    if (pix < npix) process_pixel(v, wgt, out, pix);
  }
#undef ISSUE_TILE
#else
  // Fallback: plain coalesced b128 loads (still memory-optimal).
  for (int k = 0; k < TPB; ++k) {
    const int tile = t0 + k;
    if (tile >= tend) break;
    const int pix = tile * BLOCK + tid;
    if (pix >= npix) break;
    v4f v = ((const v4f*)img)[pix];
    process_pixel(v, wgt, out, pix);
  }
#endif
}

extern "C" void kernel_launch(void* const* d_in, const int* in_sizes, int n_in,
                              void* d_out, int out_size, void* d_ws, size_t ws_size,
                              hipStream_t stream) {
  const float* img = (const float*)d_in[0];   // (16,512,512,4) f32
  const float* wgt = (const float*)d_in[1];   // (256,256,8) f32
  float* out = (float*)d_out;                 // (16,512,512,12) f32
  const int npix = in_sizes[0] / 4;           // 4,194,304
  const int num_tiles = (npix + BLOCK - 1) / BLOCK;
  const int blocks = (num_tiles + TPB - 1) / TPB;
  hipLaunchKernelGGL(warp_kernel, dim3(blocks), dim3(BLOCK), 0, stream,
                     img, wgt, out, npix);
}